// WarpedLoss_70196945486280
// MI455X (gfx1250) — compile-verified
//
#include <hip/hip_runtime.h>
#include <math.h>

#define H_DIM 2160
#define W_DIM 3840
#define NPIX (H_DIM * W_DIM)      // 8,294,400
#define NELEM (3 * NPIX)          // 24,883,200
#define NBLOCKS 2048
#define NTHREADS 256

// ix = (2*vx/(W-1) + 1)*(W/2) - 0.5  ==  vx * (W/(W-1)) - 0.5
#define SCALE_X ((float)W_DIM / (float)(W_DIM - 1))
#define SCALE_Y ((float)H_DIM / (float)(H_DIM - 1))
#define INV_NELEM (1.0f / (float)NELEM)

typedef __attribute__((ext_vector_type(2))) float v2f;
typedef __attribute__((ext_vector_type(8))) float v8f;

// occlusion indicator m(y,x) = |da + db| > 0.75 with zero padding on last row/col
__device__ __forceinline__ bool occ_m(const float* __restrict__ f0,
                                      const float* __restrict__ f1,
                                      int x, int y) {
    float da = 0.0f, db = 0.0f;
    int i = y * W_DIM + x;
    if (y < H_DIM - 1) da = f0[i + W_DIM] - f0[i];
    if (x < W_DIM - 1) db = f1[i + 1] - f1[i];
    return fabsf(da + db) > 0.75f;
}

__global__ void __launch_bounds__(NTHREADS)
warp_loss_kernel(const float* __restrict__ inp,
                 const float* __restrict__ tgt,
                 const float* __restrict__ flow,
                 float* __restrict__ partial) {
    const float* f0 = flow;          // flow[0,0,:,:]
    const float* f1 = flow + NPIX;   // flow[0,1,:,:]
    float acc = 0.0f;

    for (int i = blockIdx.x * NTHREADS + threadIdx.x; i < NPIX;
         i += NBLOCKS * NTHREADS) {
        int y = i / W_DIM;
        int x = i - y * W_DIM;

        // ---- warp coordinates (grid_sample, align_corners=False) ----
        // no divisions in the hot loop: multiply by precomputed W/(W-1), H/(H-1)
        float ix = fmaf((float)x + f0[i], SCALE_X, -0.5f);
        float iy = fmaf((float)y + f1[i], SCALE_Y, -0.5f);

        float x0f = floorf(ix), y0f = floorf(iy);
        float dx = ix - x0f, dy = iy - y0f;
        int x0 = (int)x0f, y0 = (int)y0f;
        int x1 = x0 + 1, y1 = y0 + 1;

        bool vx0 = (x0 >= 0) && (x0 < W_DIM);
        bool vx1 = (x1 >= 0) && (x1 < W_DIM);
        bool vy0 = (y0 >= 0) && (y0 < H_DIM);
        bool vy1 = (y1 >= 0) && (y1 < H_DIM);

        int cx0 = min(max(x0, 0), W_DIM - 1);
        int cx1 = min(max(x1, 0), W_DIM - 1);
        int cy0 = min(max(y0, 0), H_DIM - 1);
        int cy1 = min(max(y1, 0), H_DIM - 1);

        float w00 = (1.0f - dx) * (1.0f - dy);
        float w10 = dx * (1.0f - dy);
        float w01 = (1.0f - dx) * dy;
        float w11 = dx * dy;

        float v00 = (vx0 && vy0) ? 1.0f : 0.0f;
        float v10 = (vx1 && vy0) ? 1.0f : 0.0f;
        float v01 = (vx0 && vy1) ? 1.0f : 0.0f;
        float v11 = (vx1 && vy1) ? 1.0f : 0.0f;

        int i00 = cy0 * W_DIM + cx0;
        int i10 = cy0 * W_DIM + cx1;
        int i01 = cy1 * W_DIM + cx0;
        int i11 = cy1 * W_DIM + cx1;

        // validity mask from sampling a ones-image, thresholded
        float maskval = w00 * v00 + w10 * v10 + w01 * v01 + w11 * v11;
        float gmask = (maskval < 0.9999f) ? 0.0f : 1.0f;

        // ---- occlusion mask ----
        bool mC = occ_m(f0, f1, x, y);
        bool mU = (y > 0)         ? occ_m(f0, f1, x, y - 1) : false;
        bool mD = (y < H_DIM - 1) ? occ_m(f0, f1, x, y + 1) : false;
        bool mL = (x > 0)         ? occ_m(f0, f1, x - 1, y) : false;
        bool mR = (x < W_DIM - 1) ? occ_m(f0, f1, x + 1, y) : false;
        bool border = (x == 0) || (x == W_DIM - 1) || (y == 0) || (y == H_DIM - 1);

        bool occ0  = mC || mU || mD || mL || mR || border;  // channel 0: dilated
        bool occ12 = mC || border;                          // channels 1,2

        float fm0  = gmask * (occ0  ? 0.0f : 1.0f);
        float fm12 = gmask * (occ12 ? 0.0f : 1.0f);

        // ---- per-channel warped value and squared error ----
        #pragma unroll
        for (int c = 0; c < 3; ++c) {
            const float* tc = tgt + c * NPIX;
            float wv = w00 * (tc[i00] * v00) + w10 * (tc[i10] * v10)
                     + w01 * (tc[i01] * v01) + w11 * (tc[i11] * v11);
            float fm = (c == 0) ? fm0 : fm12;
            float d = fm * inp[c * NPIX + i] - fm * wv;
            acc += d * d;
        }
    }

    // deterministic block reduction
    __shared__ float sdata[NTHREADS];
    sdata[threadIdx.x] = acc;
    __syncthreads();
    for (int s = NTHREADS / 2; s > 0; s >>= 1) {
        if (threadIdx.x < s) sdata[threadIdx.x] += sdata[threadIdx.x + s];
        __syncthreads();
    }
    if (threadIdx.x == 0) partial[blockIdx.x] = sdata[0];
}

// Final reduction of NBLOCKS partials -> scalar mean, using
// V_WMMA_F32_16X16X4_F32 for the 64->16 row-sum stage (deterministic).
__global__ void __launch_bounds__(NTHREADS)
reduce_wmma_kernel(const float* __restrict__ partial, float* __restrict__ out) {
    __shared__ float s[NTHREADS];
    __shared__ float s2[64];
    int t = threadIdx.x;

    float a = 0.0f;
    for (int i = t; i < NBLOCKS; i += NTHREADS) a += partial[i];
    s[t] = a;
    __syncthreads();

    if (t < 64) s2[t] = s[4 * t] + s[4 * t + 1] + s[4 * t + 2] + s[4 * t + 3];
    __syncthreads();

    if (t < 32) {  // wave 0: EXEC all ones within this wave
        int lane = t;
        // A: 16x4 f32 per ISA layout (lane<16 holds K=0,1; lane>=16 holds K=2,3)
        v2f A;
        if (lane < 16) {
            A.x = s2[4 * lane + 0];
            A.y = s2[4 * lane + 1];
        } else {
            A.x = s2[4 * (lane - 16) + 2];
            A.y = s2[4 * (lane - 16) + 3];
        }
        // B: 4x16 all-ones -> D[m][n] = rowsum(A, m)
        v2f B;
        B.x = 1.0f;
        B.y = 1.0f;
        v8f C = {};
#if __has_builtin(__builtin_amdgcn_wmma_f32_16x16x4_f32)
        C = __builtin_amdgcn_wmma_f32_16x16x4_f32(
            /*neg_a=*/false, A, /*neg_b=*/false, B,
            /*c_mod=*/(short)0, C, /*reuse_a=*/false, /*reuse_b=*/false);
        // lane 0 holds D[0..7][0], lane 16 holds D[8..15][0]
        float r = C[0] + C[1] + C[2] + C[3] + C[4] + C[5] + C[6] + C[7];
#else
        // scalar fallback (should not be taken on gfx1250)
        float r = 0.0f;
        if (lane == 0)  for (int m = 0; m < 8;  ++m)
            r += s2[4*m] + s2[4*m+1] + s2[4*m+2] + s2[4*m+3];
        if (lane == 16) for (int m = 8; m < 16; ++m)
            r += s2[4*m] + s2[4*m+1] + s2[4*m+2] + s2[4*m+3];
#endif
        float other = __shfl_down(r, 16, 32);
        if (lane == 0) out[0] = (r + other) * INV_NELEM;
    }
}

extern "C" void kernel_launch(void* const* d_in, const int* in_sizes, int n_in,
                              void* d_out, int out_size, void* d_ws, size_t ws_size,
                              hipStream_t stream) {
    (void)in_sizes; (void)n_in; (void)out_size; (void)ws_size;
    const float* inp  = (const float*)d_in[0];
    const float* tgt  = (const float*)d_in[1];
    const float* flow = (const float*)d_in[2];
    float* out = (float*)d_out;
    float* partial = (float*)d_ws;  // NBLOCKS floats (8 KB)

    warp_loss_kernel<<<NBLOCKS, NTHREADS, 0, stream>>>(inp, tgt, flow, partial);
    reduce_wmma_kernel<<<1, NTHREADS, 0, stream>>>(partial, out);
}